// MaceEquivariantScorePredictionHead_83665962926838
// MI455X (gfx1250) — compile-verified
//
#include <hip/hip_runtime.h>

// MACE equivariant score head, folded form for gfx1250 (MI455X).
//
// Key observation: the reference discards s3 (silu(s2)) — TensorSquare paths
// A-D never reach the output. out depends only on path E, whose weight chain
// (W_t, W1_s, Wsv, W2, W1_v) folds into a single effective map:
//   h[n,:] = alpha * ( times[n]*qt + feat_s[n,:] @ Q )        ([N,128] GEMM)
//   out[n,m] = sum_p feat_v[n,p,m] * h[n,p]
// alpha = 1/(S*V*sqrt(V)) collects every fan-in normalization.
//
// Memory-bound: ~18 MB of reads (feat 8.4 MB + Wsv 9.4 MB) => ~0.8 us at
// 23.3 TB/s HBM. Compute is ~140 MFLOP, so we keep full f32 precision and use
// V_WMMA_F32_16X16X4_F32 for the node GEMM (exact f32 matrix path on CDNA5).

typedef __attribute__((ext_vector_type(2))) float v2f;
typedef __attribute__((ext_vector_type(8))) float v8f;

constexpr int SDIM  = 144;   // 16 time-emb scalars + 128 feature scalars
constexpr int VDIM  = 128;   // vector multiplicity
constexpr int FSC   = 128;   // scalar part of node features
constexpr int ROWF  = 512;   // FSC + 3*VDIM floats per node row
constexpr int NT    = 32;    // nodes per workgroup
constexpr int FPAD  = 132;   // LDS row pad (dwords): even (b64 align) + conflict-free

// ---------------- weight-folding kernels (all tiny, L2 resident) ------------

// M[u,v] = sum_w Wsv[u,v,w] * W2[w]        (dominant traffic: 9.4 MB of Wsv)
__global__ void k_fold_M(const float* __restrict__ Wsv, const float* __restrict__ W2,
                         float* __restrict__ M) {
  int u = blockIdx.x, v = threadIdx.x;
  const float* p = Wsv + ((size_t)u * VDIM + v) * VDIM;
  float acc = 0.f;
#pragma unroll 8
  for (int w = 0; w < VDIM; ++w) acc += p[w] * W2[w];
  M[u * VDIM + v] = acc;
}

// K[j,v] = sum_u W1_s[j,u] * M[u,v]
__global__ void k_fold_K(const float* __restrict__ W1s, const float* __restrict__ M,
                         float* __restrict__ K) {
  int j = blockIdx.x, v = threadIdx.x;
  float acc = 0.f;
  for (int u = 0; u < SDIM; ++u) acc += W1s[j * SDIM + u] * M[u * VDIM + v];
  K[j * VDIM + v] = acc;
}

// kt[v] = sum_{j<16} W_t[j] * K[j,v]   (time-embedding rows pre-contracted)
__global__ void k_fold_kt(const float* __restrict__ Wt, const float* __restrict__ K,
                          float* __restrict__ kt) {
  int v = threadIdx.x;
  float acc = 0.f;
#pragma unroll
  for (int j = 0; j < 16; ++j) acc += Wt[j] * K[j * VDIM + v];
  kt[v] = acc;
}

// QT[p,j] = alpha * sum_v K[16+j,v] * W1_v[p,v]   (stored K-major => b64 B frags)
__global__ void k_fold_QT(const float* __restrict__ K, const float* __restrict__ W1v,
                          float* __restrict__ QT) {
  int p = blockIdx.x, j = threadIdx.x;
  const float alpha = 1.0f / (144.0f * 128.0f * sqrtf(128.0f));
  float acc = 0.f;
  for (int v = 0; v < VDIM; ++v) acc += K[(16 + j) * VDIM + v] * W1v[p * VDIM + v];
  QT[p * VDIM + j] = alpha * acc;
}

// qt[p] = alpha * sum_v kt[v] * W1_v[p,v]
__global__ void k_fold_qt(const float* __restrict__ kt, const float* __restrict__ W1v,
                          float* __restrict__ qt) {
  int p = threadIdx.x;
  const float alpha = 1.0f / (144.0f * 128.0f * sqrtf(128.0f));
  float acc = 0.f;
  for (int v = 0; v < VDIM; ++v) acc += kt[v] * W1v[p * VDIM + v];
  qt[p] = alpha * acc;
}

// ---------------- main node kernel ------------------------------------------
// 256 threads = 8 wave32 per 32-node tile. Wave w owns output columns
// p in [16w, 16w+16). h-tile via 64x v_wmma_f32_16x16x4_f32 (2 row tiles x
// 32 k-steps), then per-lane contraction with the 1o vector features.
__global__ __launch_bounds__(256) void k_main(const float* __restrict__ feat,
                                              const float* __restrict__ times,
                                              const float* __restrict__ QT,
                                              const float* __restrict__ qt,
                                              float* __restrict__ out, int N) {
  __shared__ float Fsm[NT * FPAD];   // staged feat scalars, padded rows
  __shared__ float qts[VDIM];
  __shared__ float tms[NT];
  __shared__ float accs[NT * 3];

  const int t = threadIdx.x;
  const int base = blockIdx.x * NT;

  // Stage the 32x128 scalar block (coalesced global, conflict-free LDS).
  for (int idx = t; idx < NT * FSC; idx += 256) {
    int n = idx >> 7, j = idx & 127;
    float val = 0.f;
    if (base + n < N) val = feat[(size_t)(base + n) * ROWF + j];
    Fsm[n * FPAD + j] = val;
  }
  if (t < VDIM) qts[t] = qt[t];
  if (t < NT)   tms[t] = (base + t < N) ? times[base + t] : 0.f;
  if (t < NT * 3) accs[t] = 0.f;
  __syncthreads();

  const int w    = t >> 5;      // wave 0..7 -> column block
  const int l    = t & 31;      // lane
  const int half = l >> 4;      // k-half within the 16x16x4 fragment
  const int nl   = l & 15;
  const int pcol = w * 16 + nl; // output column p this lane carries

  // Hoist this wave's B fragments: Q^T row pcol, 32 k-steps of 2 floats.
  const float* qrow = QT + (size_t)pcol * VDIM + 2 * half;
  v2f breg[32];
#pragma unroll
  for (int kb = 0; kb < 32; ++kb) breg[kb] = *(const v2f*)(qrow + 4 * kb);

  v8f c0 = {};  // nodes 0..15 of tile
  v8f c1 = {};  // nodes 16..31 of tile
#pragma unroll
  for (int kb = 0; kb < 32; ++kb) {
    const int ko = 4 * kb + 2 * half;
    v2f a0 = *(const v2f*)(&Fsm[nl * FPAD + ko]);          // A[M=nl,   k..k+1]
    v2f a1 = *(const v2f*)(&Fsm[(16 + nl) * FPAD + ko]);   // A[M=16+nl,k..k+1]
    c0 = __builtin_amdgcn_wmma_f32_16x16x4_f32(false, a0, false, breg[kb],
                                               (short)0, c0, false, false);
    c1 = __builtin_amdgcn_wmma_f32_16x16x4_f32(false, a1, false, breg[kb],
                                               (short)0, c1, false, false);
  }

  // Finalize: h = C + times*qt, contract with vector features, reduce over p.
  const float qv = qts[pcol];
#pragma unroll
  for (int rt = 0; rt < 2; ++rt) {
    v8f c = rt ? c1 : c0;
#pragma unroll
    for (int r = 0; r < 8; ++r) {
      const int n = rt * 16 + half * 8 + r;   // node within tile (D-layout M)
      const float h = c[r] + tms[n] * qv;
      float px = 0.f, py = 0.f, pz = 0.f;
      if (base + n < N) {
        const float* fv = feat + (size_t)(base + n) * ROWF + FSC + 3 * pcol;
        px = h * fv[0]; py = h * fv[1]; pz = h * fv[2];
      }
#pragma unroll
      for (int m = 8; m; m >>= 1) {           // tree-reduce the 16 p-lanes
        px += __shfl_xor(px, m, 16);
        py += __shfl_xor(py, m, 16);
        pz += __shfl_xor(pz, m, 16);
      }
      if (nl == 0) {                           // one lane per (half, node)
        atomicAdd(&accs[n * 3 + 0], px);
        atomicAdd(&accs[n * 3 + 1], py);
        atomicAdd(&accs[n * 3 + 2], pz);
      }
    }
  }
  __syncthreads();

  if (t < NT * 3) {
    int n = t / 3;
    if (base + n < N) out[(size_t)(base + n) * 3 + (t - n * 3)] = accs[t];
  }
}

// ---------------- launcher ---------------------------------------------------
extern "C" void kernel_launch(void* const* d_in, const int* in_sizes, int n_in,
                              void* d_out, int out_size, void* d_ws, size_t ws_size,
                              hipStream_t stream) {
  (void)n_in; (void)out_size; (void)ws_size;
  const float* feat  = (const float*)d_in[0];  // [N,512]
  const float* times = (const float*)d_in[1];  // [N,1]
  const float* Wt    = (const float*)d_in[2];  // [1,16]
  const float* W1s   = (const float*)d_in[3];  // [144,144]
  const float* W1v   = (const float*)d_in[4];  // [128,128]
  // d_in[5] = Wss, d_in[7] = Wvv: dead w.r.t. the returned output (s3 deleted)
  const float* Wsv   = (const float*)d_in[6];  // [144,128,128]
  const float* W2    = (const float*)d_in[8];  // [128,1]
  float* out = (float*)d_out;

  const int N = in_sizes[0] / ROWF;

  // workspace layout (floats): needs ~214 KB
  float* ws = (float*)d_ws;
  float* M  = ws;                    // 144*128
  float* K  = M + SDIM * VDIM;       // 144*128
  float* kt = K + SDIM * VDIM;       // 128
  float* QT = kt + VDIM;             // 128*128
  float* qt = QT + VDIM * VDIM;      // 128

  hipLaunchKernelGGL(k_fold_M,  dim3(SDIM), dim3(VDIM), 0, stream, Wsv, W2, M);
  hipLaunchKernelGGL(k_fold_K,  dim3(SDIM), dim3(VDIM), 0, stream, W1s, M, K);
  hipLaunchKernelGGL(k_fold_kt, dim3(1),    dim3(VDIM), 0, stream, Wt, K, kt);
  hipLaunchKernelGGL(k_fold_QT, dim3(VDIM), dim3(VDIM), 0, stream, K, W1v, QT);
  hipLaunchKernelGGL(k_fold_qt, dim3(1),    dim3(VDIM), 0, stream, kt, W1v, qt);

  const int blocks = (N + NT - 1) / NT;
  hipLaunchKernelGGL(k_main, dim3(blocks), dim3(256), 0, stream,
                     feat, times, QT, qt, out, N);
}